// CorrLSTM_depth_29609504538949
// MI455X (gfx1250) — compile-verified
//
#include <hip/hip_runtime.h>

#define PS   7
#define PS2  49
#define NLEV 6
#define KC   16          // channels staged per round in corr kernel

typedef __attribute__((ext_vector_type(2))) float v2f;
typedef __attribute__((ext_vector_type(8))) float v8f;

#if defined(__HIP_DEVICE_COMPILE__)
#if !__has_builtin(__builtin_amdgcn_wmma_f32_16x16x4_f32)
#error "missing __builtin_amdgcn_wmma_f32_16x16x4_f32 on this toolchain (device pass)"
#endif
#endif

__device__ __forceinline__ v8f wmma4(v2f a, v2f b, v8f c) {
  // D = A(16x4 f32) * B(4x16 f32) + C(16x16 f32)
  return __builtin_amdgcn_wmma_f32_16x16x4_f32(false, a, false, b, (short)0, c,
                                               false, false);
}

__device__ __forceinline__ int imin(int a, int b) { return a < b ? a : b; }

// ---------------------------------------------------------------------------
// Correlation as banded QK^T. Block = 1 wave, tile (b, y, x0..x0+15).
// Stage KC=16 channels of f1 (queries) and the 7-row/22-col f2 key band into
// LDS (zero padded there), then run an unconditional ds_load_b64 + WMMA loop:
// 4 K-chunks x 7 dy x 2 key tiles per round.
// ---------------------------------------------------------------------------
__global__ __launch_bounds__(32)
void corr_wmma_kernel(const float* __restrict__ f1, const float* __restrict__ f2,
                      float* __restrict__ corr_out, int C, int H) {
  const int W    = H;
  const int x0   = blockIdx.x * 16;
  const int y    = blockIdx.y;
  const int b    = blockIdx.z;
  const int lane = threadIdx.x;
  const int m    = lane & 15;
  const int half = lane >> 4;
  const int kb   = half * 2;

  const size_t plane = (size_t)H * W;
  const float* f1b = f1 + (size_t)b * C * plane;
  const float* f2b = f2 + (size_t)b * C * plane;

  __shared__ float sf1[16 * 18];          // [px][ch pad 18]
  __shared__ float sf2[7 * 32 * 18];      // [row][col][ch pad 18]

  v8f acc[PS][2];
#pragma unroll
  for (int dy = 0; dy < PS; ++dy)
#pragma unroll
    for (int t = 0; t < 2; ++t) {
      v8f z = {};
      acc[dy][t] = z;
    }

  for (int c0 = 0; c0 < C; c0 += KC) {
    // ---- stage f1 tile: 16 ch x 16 px, coalesced over px ----
#pragma unroll
    for (int i = 0; i < 8; ++i) {
      const int idx = i * 32 + lane;      // 0..255
      const int ch  = idx >> 4;
      const int px  = idx & 15;
      float v = 0.f;
      if (x0 + px < W)
        v = f1b[(size_t)(c0 + ch) * plane + (size_t)y * W + (x0 + px)];
      sf1[px * 18 + ch] = v;
    }
    // ---- stage f2 band: 7 rows x 32 cols x 16 ch, coalesced over col ----
    for (int i = 0; i < 112; ++i) {       // 7*32*16 / 32
      const int idx  = i * 32 + lane;
      const int col  = idx & 31;
      const int rest = idx >> 5;
      const int ch   = rest & 15;
      const int row  = rest >> 4;         // 0..6
      const int yk   = y + row - PS / 2;
      const int xc   = x0 - PS / 2 + col;
      float v = 0.f;
      if (col < 22 && yk >= 0 && yk < H && xc >= 0 && xc < W)
        v = f2b[(size_t)(c0 + ch) * plane + (size_t)yk * W + xc];
      sf2[(row * 32 + col) * 18 + ch] = v;
    }
    __syncthreads();

    // ---- WMMA: all operands from LDS, unconditional b64 reads ----
#pragma unroll
    for (int cc = 0; cc < KC; cc += 4) {
      const v2f a = *(const v2f*)&sf1[m * 18 + cc + kb];
#pragma unroll
      for (int dy = 0; dy < PS; ++dy)
#pragma unroll
        for (int t = 0; t < 2; ++t) {
          const v2f bf =
              *(const v2f*)&sf2[(dy * 32 + t * 16 + m) * 18 + cc + kb];
          acc[dy][t] = wmma4(a, bf, acc[dy][t]);
        }
    }
    __syncthreads();
  }

  // Extract band: lane holds (M = r + 8*half, j = t*16 + m); dx = j - M.
#pragma unroll
  for (int dy = 0; dy < PS; ++dy)
#pragma unroll
    for (int t = 0; t < 2; ++t)
#pragma unroll
      for (int r = 0; r < 8; ++r) {
        const int mm = r + 8 * half;
        const int j  = t * 16 + m;
        const int dx = j - mm;
        if (dx >= 0 && dx < PS) {
          const int x = x0 + mm;
          if (x < W) {
            float v = acc[dy][t][r];
            v = (v > 0.f) ? v : 0.01f * v;   // leaky relu
            corr_out[((size_t)(b * PS2 + dy * PS + dx)) * plane +
                     (size_t)y * W + x] = v;
          }
        }
      }
}

// ---------------------------------------------------------------------------
// Weight repack: w[196][98][3][3] -> wp[tap(9)][ic(100)][oc(208)], zero padded.
// Makes conv A-fragment loads coalesced and unconditional.
// ---------------------------------------------------------------------------
__global__ __launch_bounds__(256)
void repack_w_kernel(const float* __restrict__ w, float* __restrict__ wp) {
  const int idx = blockIdx.x * 256 + threadIdx.x;
  if (idx >= 9 * 100 * 208) return;
  const int oc   = idx % 208;
  const int rest = idx / 208;
  const int ic   = rest % 100;
  const int tap  = rest / 100;
  float v = 0.f;
  if (oc < 4 * PS2 && ic < 2 * PS2) v = w[((size_t)oc * 98 + ic) * 9 + tap];
  wp[idx] = v;
}

// ---------------------------------------------------------------------------
// 3x3 conv (98 -> 196 ch) as implicit GEMM + LSTM gates. Block = 1 wave.
// sx is channel-fastest with zero padding -> B frag = one ds_load_b64.
// A loads are unconditional (REPACK: coalesced from wp; else clamped direct).
// 2 oc-tiles per pass share each B fragment.
// ---------------------------------------------------------------------------
template <bool REPACK>
__global__ __launch_bounds__(32)
void conv_lstm_wmma_kernel(const float* __restrict__ cf,
                           const float* __restrict__ hin,
                           const float* __restrict__ cin,
                           const float* __restrict__ wsrc,
                           const float* __restrict__ bias,
                           float* __restrict__ hout,
                           float* __restrict__ cout_, int H) {
  const int W    = H;
  const int x0   = blockIdx.x * 16;
  const int y    = blockIdx.y;
  const int b    = blockIdx.z;
  const int lane = threadIdx.x;
  const int n    = lane & 15;
  const int half = lane >> 4;
  const int kb   = half * 2;

  __shared__ float sx[3 * 18 * 100];   // [row][col][ch pad 100], zero padded
  __shared__ float co[208 * 16];       // conv outputs [oc][pixel]

  const size_t plane = (size_t)H * W;

  // ---- stage input slab (coalesced over col; ch 98..99 & OOB -> 0) ----
  for (int idx = lane; idx < 100 * 54; idx += 32) {
    const int ic  = idx / 54;
    const int rem = idx - ic * 54;
    const int row = rem / 18;
    const int col = rem - row * 18;
    const int yy  = y + row - 1;
    const int xx  = x0 + col - 1;
    float v = 0.f;
    if (ic < 2 * PS2 && yy >= 0 && yy < H && xx >= 0 && xx < W) {
      v = (ic < PS2)
              ? cf[((size_t)(b * PS2 + ic)) * plane + (size_t)yy * W + xx]
              : hin[((size_t)(b * PS2 + ic - PS2)) * plane + (size_t)yy * W + xx];
    }
    sx[(row * 18 + col) * 100 + ic] = v;
  }
  __syncthreads();

  for (int tp = 0; tp < 7; ++tp) {           // 13 oc tiles, blocked by 2
    const int t0  = 2 * tp, t1 = 2 * tp + 1; // t1 == 13 is padding work
    const int oc0 = t0 * 16 + n;
    const int oc1 = t1 * 16 + n;
    const int oc0c = imin(oc0, REPACK ? 207 : 195);
    const int oc1c = imin(oc1, REPACK ? 207 : 195);
    v8f acc0 = {}, acc1 = {};
#pragma unroll
    for (int ky = 0; ky < 3; ++ky) {
#pragma unroll
      for (int kx = 0; kx < 3; ++kx) {
        const int tap = ky * 3 + kx;
        for (int c0 = 0; c0 < 100; c0 += 4) {
          const int ic0 = c0 + kb;           // even, <= 98
          const v2f bf = *(const v2f*)&sx[(ky * 18 + n + kx) * 100 + ic0];
          v2f a0, a1;
          if (REPACK) {
            const float* p = wsrc + ((size_t)tap * 100 + ic0) * 208;
            a0.x = p[oc0c];       a0.y = p[208 + oc0c];
            a1.x = p[oc1c];       a1.y = p[208 + oc1c];
          } else {
            const int i0 = imin(ic0, 97), i1 = imin(ic0 + 1, 97);
            a0.x = wsrc[((size_t)oc0c * 98 + i0) * 9 + tap];
            a0.y = wsrc[((size_t)oc0c * 98 + i1) * 9 + tap];
            a1.x = wsrc[((size_t)oc1c * 98 + i0) * 9 + tap];
            a1.y = wsrc[((size_t)oc1c * 98 + i1) * 9 + tap];
          }
          acc0 = wmma4(a0, bf, acc0);
          acc1 = wmma4(a1, bf, acc1);
        }
      }
    }
#pragma unroll
    for (int r = 0; r < 8; ++r) {
      const int oca = t0 * 16 + r + 8 * half;
      const int ocb = t1 * 16 + r + 8 * half;
      if (oca < 4 * PS2) co[oca * 16 + n] = acc0[r] + bias[oca];
      if (ocb < 4 * PS2) co[ocb * 16 + n] = acc1[r] + bias[ocb];
    }
  }
  __syncthreads();

  // ---- LSTM gates: i | f | o | g along oc ----
  for (int idx = lane; idx < PS2 * 16; idx += 32) {
    const int gc = idx >> 4;
    const int nn = idx & 15;
    const int x  = x0 + nn;
    if (x >= W) continue;
    float ig = co[(gc) * 16 + nn];
    float fg = co[(gc + PS2) * 16 + nn];
    float og = co[(gc + 2 * PS2) * 16 + nn];
    float gg = co[(gc + 3 * PS2) * 16 + nn];
    ig = 1.f / (1.f + __expf(-ig));
    fg = 1.f / (1.f + __expf(-fg));
    og = 1.f / (1.f + __expf(-og));
    {
      const float e = __expf(-2.f * fabsf(gg));
      const float t = (1.f - e) / (1.f + e);
      gg = (gg >= 0.f) ? t : -t;
    }
    const size_t oidx = ((size_t)(b * PS2 + gc)) * plane + (size_t)y * W + x;
    const float cprev = cin[oidx];
    const float cn    = fg * cprev + ig * gg;
    float th;
    {
      const float e = __expf(-2.f * fabsf(cn));
      const float t = (1.f - e) / (1.f + e);
      th = (cn >= 0.f) ? t : -t;
    }
    hout[oidx]  = og * th;
    cout_[oidx] = cn;
  }
}

// ---------------------------------------------------------------------------
extern "C" void kernel_launch(void* const* d_in, const int* in_sizes, int n_in,
                              void* d_out, int out_size, void* d_ws,
                              size_t ws_size, hipStream_t stream) {
  (void)in_sizes; (void)n_in; (void)out_size;
  static const int CH[NLEV] = {512, 1024, 512, 256, 256, 256};
  static const int HH[NLEV] = {64, 32, 16, 8, 4, 2};

  size_t off[NLEV], G = 0;
  for (int i = 0; i < NLEV; ++i) {
    off[i] = G;
    G += (size_t)8 * PS2 * HH[i] * HH[i];
  }

  const size_t repack_bytes = (size_t)9 * 100 * 208 * sizeof(float);
  const bool use_repack = (ws_size >= repack_bytes) && (d_ws != nullptr);
  float* wp = (float*)d_ws;

  float* out = (float*)d_out;
  for (int lev = 0; lev < NLEV; ++lev) {
    const float* x    = (const float*)d_in[6 * lev + 0];
    const float* xp   = (const float*)d_in[6 * lev + 1];
    const float* h    = (const float*)d_in[6 * lev + 2];
    const float* c    = (const float*)d_in[6 * lev + 3];
    const float* w    = (const float*)d_in[6 * lev + 4];
    const float* bias = (const float*)d_in[6 * lev + 5];
    const int H = HH[lev], C = CH[lev];
    const int tiles = (H + 15) / 16;

    float* hn  = out + off[lev];          // h0..h5
    float* cn  = out + G + off[lev];      // c0..c5
    float* cfp = out + 2 * G + off[lev];  // cf0..cf5

    dim3 grid(tiles, H, 8);
    corr_wmma_kernel<<<grid, 32, 0, stream>>>(x, xp, cfp, C, H);
    if (use_repack) {
      repack_w_kernel<<<(9 * 100 * 208 + 255) / 256, 256, 0, stream>>>(w, wp);
      conv_lstm_wmma_kernel<true><<<grid, 32, 0, stream>>>(cfp, h, c, wp, bias,
                                                           hn, cn, H);
    } else {
      conv_lstm_wmma_kernel<false><<<grid, 32, 0, stream>>>(cfp, h, c, w, bias,
                                                            hn, cn, H);
    }
  }
}